// GumbelVectorQuantizer_15582141350646
// MI455X (gfx1250) — compile-verified
//
#include <hip/hip_runtime.h>

// Gumbel vector quantizer, fused on gfx1250:
//   f32 WMMA GEMM (16x16x4) with LDS-staged, double-buffered async W chunks
//   -> bias -> dual softmax -> first-index argmax one-hot -> codebook gather;
//   perplexity via global f32 atomics.
// Sizes: B=16, L=2048, N=32768 tokens, D1=512, G=2, V=320, DG=384.

typedef __attribute__((ext_vector_type(2))) float v2f;
typedef __attribute__((ext_vector_type(8))) float v8f;
typedef __attribute__((ext_vector_type(4))) int   v4i;

#define D1   512
#define NV   320
#define GNUM 2
#define NOUT 640           // G*V
#define DGV  384           // per-group code vector dim
#define MT   64            // tokens per block (512 blocks)
#define KC   32            // K-chunk of W staged in LDS
#define AP   516           // A tile padded row stride (words)
#define WP   36            // W chunk padded row stride (words): conflict-free b64 frags
#define LP   644           // logits padded row stride (words)

#define W_BYTES    (NOUT * WP * 4)            // 92160
#define A_BYTES    (MT * AP * 4)              // 132096
#define SMEM_BYTES (2 * W_BYTES + A_BYTES)    // 316416  (< 320KB/WGP)
// logits tile (MT*LP*4 = 164864) reuses the same region after the GEMM barrier

#if __has_builtin(__builtin_amdgcn_global_load_async_to_lds_b128) && \
    __has_builtin(__builtin_amdgcn_s_wait_asynccnt)
#define HAS_ASYNC_LDS 1
#else
#define HAS_ASYNC_LDS 0
#endif

__device__ __forceinline__ void cp16_to_lds(void* lds, const void* g) {
#if HAS_ASYNC_LDS
  // per-lane 16B DMA: copies 16 bytes from global address g to LDS address lds; tracked by ASYNCcnt
  __builtin_amdgcn_global_load_async_to_lds_b128(
      (__attribute__((address_space(1))) v4i*)g,
      (__attribute__((address_space(3))) v4i*)lds, 0, 0);
#else
  *(float4*)lds = *(const float4*)g;
#endif
}
__device__ __forceinline__ void lds_cp_wait() {
#if HAS_ASYNC_LDS
  __builtin_amdgcn_s_wait_asynccnt(0);
#endif
}

__device__ __forceinline__ float wred_max(float v) {
#pragma unroll
  for (int o = 16; o > 0; o >>= 1) v = fmaxf(v, __shfl_xor(v, o, 32));
  return v;
}
__device__ __forceinline__ float wred_sum(float v) {
#pragma unroll
  for (int o = 16; o > 0; o >>= 1) v += __shfl_xor(v, o, 32);
  return v;
}

__global__ void gvq_zero_kernel(float* __restrict__ p, int n) {
  int i = blockIdx.x * blockDim.x + threadIdx.x;
  if (i < n) p[i] = 0.0f;
}

__global__ __launch_bounds__(256) void gvq_main_kernel(
    const float* __restrict__ hidden,    // (32768, 512)
    const float* __restrict__ W,         // (640, 512) row-major (O, K)
    const float* __restrict__ bias,      // (640,)
    const float* __restrict__ codebook,  // (2, 320, 384)
    const float* __restrict__ gumbel,    // (65536, 320)
    float* __restrict__ out_cv,          // (32768, 768)
    float* __restrict__ out_perp)        // (2048, 2, 320), pre-zeroed
{
  extern __shared__ char smem[];
  float* Wbuf0 = (float*)smem;                       // 90KB ping
  float* Wbuf1 = (float*)(smem + W_BYTES);           // 90KB pong
  float* Albs  = (float*)(smem + 2 * W_BYTES);       // 129KB A tile
  float* Llds  = (float*)smem;                       // reused post-GEMM

  const int n0   = blockIdx.x * MT;
  const int tid  = threadIdx.x;
  const int wave = tid >> 5;          // 0..7
  const int lane = tid & 31;

  // ---- Issue A tile copy (64 x 512 f32 -> padded LDS), 16B per lane ----
#pragma unroll 4
  for (int j = tid; j < MT * D1 / 4; j += 256) {     // 8192 float4, 32/thread
    const int row = j >> 7, cw = j & 127;
    cp16_to_lds(&Albs[row * AP + cw * 4],
                hidden + (size_t)(n0 + row) * D1 + cw * 4);
  }
  // ---- Issue W chunk 0 (640 x 32 f32) ----
#pragma unroll 4
  for (int j = tid; j < NOUT * KC / 4; j += 256) {   // 5120 float4, 20/thread
    const int col = j >> 3, q = j & 7;
    cp16_to_lds(&Wbuf0[col * WP + q * 4],
                W + (size_t)col * D1 + q * 4);
  }

  // ---- GEMM: wave owns 16 rows x 320 cols (one group) = 20 accum tiles ----
  v8f acc[20];
#pragma unroll
  for (int t = 0; t < 20; ++t) { v8f z = {}; acc[t] = z; }

  const int m     = lane & 15;        // M-in-subtile (A) == N-in-tile (B)
  const int kh    = lane >> 4;        // K-half selector
  const int g     = wave >> 2;        // group: cols [g*320, g*320+320)
  const int ms    = wave & 3;         // M-subtile: rows [ms*16, ms*16+16)
  const int cbase = g * NV;
  const int arow  = ms * 16 + m;

  for (int c = 0; c < D1 / KC; ++c) {                 // 16 K-chunks
    lds_cp_wait();                                    // own async issues done
    __syncthreads();                                  // all waves' data visible
    if (c + 1 < D1 / KC) {                            // prefetch next chunk
      float* nb = ((c + 1) & 1) ? Wbuf1 : Wbuf0;
#pragma unroll 4
      for (int j = tid; j < NOUT * KC / 4; j += 256) {
        const int col = j >> 3, q = j & 7;
        cp16_to_lds(&nb[col * WP + q * 4],
                    W + (size_t)col * D1 + (c + 1) * KC + q * 4);
      }
    }
    const float* wb = (c & 1) ? Wbuf1 : Wbuf0;
    const int kbase = c * KC;
    for (int kk = 0; kk < KC; kk += 4) {
      // A frag 16x4: lane holds A[arow][k + 2*kh + {0,1}] (ds_load_b64)
      v2f a = *(const v2f*)&Albs[arow * AP + kbase + kk + 2 * kh];
#pragma unroll
      for (int t = 0; t < 20; ++t) {
        // B frag 4x16 from LDS: W[col][k + 2*kh + {0,1}], conflict-free
        const int col = cbase + t * 16 + m;
        v2f b = *(const v2f*)&wb[col * WP + kk + 2 * kh];
        acc[t] = __builtin_amdgcn_wmma_f32_16x16x4_f32(
            false, a, false, b, (short)0, acc[t], false, false);
      }
    }
  }
  __syncthreads();   // done reading A/W LDS before logits tile overwrites it

  // ---- Spill logits (+bias): C layout VGPR r -> M = r + 8*(lane>=16) ----
#pragma unroll
  for (int t = 0; t < 20; ++t) {
    const int col = cbase + t * 16 + m;
    const float bv = bias[col];
#pragma unroll
    for (int r = 0; r < 8; ++r) {
      const int row = ms * 16 + r + kh * 8;
      Llds[row * LP + col] = acc[t][r] + bv;
    }
  }
  __syncthreads();

  // ---- Per (token, group) row: gumbel softmax, perplexity, gather ----
  // 128 rows; 16 per wave; 320 cols = 10 per lane.
  for (int rg = wave; rg < 2 * MT; rg += 8) {
    const int tl = rg >> 1;
    const int gg = rg & 1;
    const int n  = n0 + tl;
    const int l  = n & 2047;                 // n = b*2048 + l

    const float* lrow = &Llds[tl * LP + gg * NV];
    const float* grow = gumbel + ((size_t)(2 * n + gg)) * NV;

    float zv[10], lv[10];
    float zbest = -3.402823466e38f, lmax = -3.402823466e38f;
    int   ibest = 0;
#pragma unroll
    for (int j = 0; j < 10; ++j) {
      const int c   = lane + 32 * j;
      const float lg = lrow[c];
      const float z  = (lg + grow[c]) * 0.5f;      // (logit + gumbel)/TAU
      zv[j] = z; lv[j] = lg;
      if (z > zbest) { zbest = z; ibest = c; }     // strict > keeps first idx
      lmax = fmaxf(lmax, lg);
    }
    // wave argmax, tie-break smallest index (matches jnp.argmax)
#pragma unroll
    for (int o = 16; o > 0; o >>= 1) {
      const float ov = __shfl_xor(zbest, o, 32);
      const int   oi = __shfl_xor(ibest, o, 32);
      if (ov > zbest || (ov == zbest && oi < ibest)) { zbest = ov; ibest = oi; }
    }
    const float zmax = zbest;
    lmax = wred_max(lmax);

    float zs = 0.0f, ls = 0.0f;
    float e2[10];
#pragma unroll
    for (int j = 0; j < 10; ++j) {
      zs += __expf(zv[j] - zmax);
      e2[j] = __expf(lv[j] - lmax);
      ls += e2[j];
    }
    zs = wred_sum(zs);
    ls = wred_sum(ls);

    // perplexity[l,g,v] += softmax(logits)[v] / B
    const float inv_ls_b = 1.0f / (ls * 16.0f);
    float* prow = out_perp + ((size_t)l * GNUM + gg) * NV;
#pragma unroll
    for (int j = 0; j < 10; ++j)
      atomicAdd(&prow[lane + 32 * j], e2[j] * inv_ls_b);

    // straight-through hot value: (1 + y_soft[amax]) - y_soft[amax]
    const float s = 1.0f / zs;
    const float p = (1.0f + s) - s;

    const float* cb  = codebook + ((size_t)gg * NV + ibest) * DGV;
    float*       cvr = out_cv + (size_t)n * (GNUM * DGV) + (size_t)gg * DGV;
#pragma unroll
    for (int j = 0; j < 12; ++j)
      cvr[lane + 32 * j] = cb[lane + 32 * j] * p;
  }
}

extern "C" void kernel_launch(void* const* d_in, const int* in_sizes, int n_in,
                              void* d_out, int out_size, void* d_ws, size_t ws_size,
                              hipStream_t stream) {
  (void)in_sizes; (void)n_in; (void)out_size; (void)d_ws; (void)ws_size;
  const float* hidden   = (const float*)d_in[0];   // (16,2048,512)
  const float* W        = (const float*)d_in[1];   // (640,512)
  const float* bias     = (const float*)d_in[2];   // (640,)
  const float* codebook = (const float*)d_in[3];   // (1,2,320,384)
  const float* gumbel   = (const float*)d_in[4];   // (65536,320)

  float* out_cv   = (float*)d_out;                      // 32768*768 floats
  float* out_perp = out_cv + (size_t)32768 * 768;       // 2048*2*320 floats

  const int np = 2048 * GNUM * NV;
  gvq_zero_kernel<<<(np + 255) / 256, 256, 0, stream>>>(out_perp, np);
  gvq_main_kernel<<<32768 / MT, 256, SMEM_BYTES, stream>>>(
      hidden, W, bias, codebook, gumbel, out_cv, out_perp);
}